// abstract_SPU_transformer_17033840296084
// MI455X (gfx1250) — compile-verified
//
#include <hip/hip_runtime.h>
#include <hip/hip_bf16.h>
#include <stdint.h>

// ---------------------------------------------------------------------------
// SPU abstract-transformer, elementwise over FC intervals.
//   out[0:n)   = lower_out
//   out[n:2n)  = upper_out
// Memory floor: 32 B/elem * 8.4M = 268 MB  -> ~11.5 us @ 23.3 TB/s.
// Compute is dominated by the case-C bisection; we early-exit it with a
// wave32 ballot and use raw v_exp2/v_rcp/v_sin transcendentals.
// lambda (16B/elem) is streamed via global_load_async_to_lds_b128 and waited
// with s_wait_asynccnt AFTER the bisection loop for full overlap.
// ---------------------------------------------------------------------------

#if defined(__HIP_DEVICE_COMPILE__) && __has_builtin(__builtin_amdgcn_global_load_async_to_lds_b128)
#define SPU_ASYNC_LDS 1
#endif

#define SPU_LOG2E   1.4426950408889634f
#define SPU_INV_2PI 0.15915494309189535f

__device__ __forceinline__ float fsig(float x) {
  // sigmoid(x) = 1 / (1 + 2^(-x*log2(e)))  -> v_exp_f32 + v_rcp_f32
  return __builtin_amdgcn_rcpf(1.0f + __builtin_amdgcn_exp2f(-x * SPU_LOG2E));
}

__device__ __forceinline__ float flink(float x) {
  // link(x) = sin(x)/2 + 0.5 ; v_sin_f32 takes revolutions (x / 2pi)
  return fmaf(0.5f, __builtin_amdgcn_sinf(x * SPU_INV_2PI), 0.5f);
}

__device__ __forceinline__ bool wave_any(bool p) {
#if __has_builtin(__builtin_amdgcn_ballot_w32)
  return __builtin_amdgcn_ballot_w32(p) != 0u;   // wave32 on gfx1250
#else
  return true;
#endif
}

__global__ __launch_bounds__(256) void spu_transformer_kernel(
    const float*  __restrict__ lower,
    const float*  __restrict__ upper,
    const float4* __restrict__ lambda,
    float*        __restrict__ out,
    int n) {
  __shared__ float4 lam_tile[256];
  const int tid = threadIdx.x;
  const int i   = blockIdx.x * 256 + tid;
  const int j   = (i < n) ? i : (n - 1);   // clamp so full wave can issue loads

  // ---- kick off async lambda fetch (ASYNCcnt-tracked); wait much later ----
#if defined(SPU_ASYNC_LDS)
  {
    // Builtin signature (per hipcc diagnostic): (v4i AS1*, v4i AS3*, Ii, Ii)
    typedef int v4i_t __attribute__((vector_size(4 * sizeof(int))));
    typedef __attribute__((address_space(1))) v4i_t gv4i_t;
    typedef __attribute__((address_space(3))) v4i_t lv4i_t;
    gv4i_t* gsrc = (gv4i_t*)(uintptr_t)(const void*)(lambda + j);
    lv4i_t* ldst = (lv4i_t*)(uint32_t)(uintptr_t)(void*)(&lam_tile[tid]);
    __builtin_amdgcn_global_load_async_to_lds_b128(gsrc, ldst, 0, 0);
  }
#else
  lam_tile[tid] = lambda[j];
#endif

  // ---- scalar inputs (streamed once; NT so 268MB stream skips L2 rinse) ----
  float lo = __builtin_nontemporal_load(lower + j);
  float up = __builtin_nontemporal_load(upper + j);
  if (up == lo) up += 1e-8f;
  const float width = up - lo;
  const float up2m  = fmaf(up, up, -0.5f);          // upper^2 - 0.5

  const bool mask_a = (lo >= 0.0f);
  const bool mask_b = (up <= 0.0f);
  const bool mask_c = (lo < 0.0f) && (up > 0.0f);

  // ---- case-C shared pieces -----------------------------------------------
  const float sL         = fsig(lo);
  const float slope_tan  = -sL * (1.0f - sL);
  const float slope_line = (up2m + sL) * __builtin_amdgcn_rcpf(width);
  const bool  use_line   = slope_line >= slope_tan;

  // ---- bisection: only lanes with (mask_c && !use_line); ~13 real iters ----
  const bool need = mask_c && !use_line;
  float st = need ? lo : 0.0f;   // inactive lanes: st==ed==0 -> converged
  float ed = 0.0f;
  if (wave_any(need)) {
#pragma unroll 1
    for (int it = 0; it < 40; ++it) {
      const bool active = (ed - st) > 0.001f;
      if (!wave_any(active)) break;              // wave32 ballot early-exit
      const float mid = (st + ed) * 0.5f;
      const float s   = fsig(mid);
      const float t   = fmaf(-s, s, s);          // s*(1-s)
      const float y   = fmaf(t, mid - up, -s);   // -s(1-s)(up-mid) - s
      const bool  gr  = y > up2m;
      st = (active &&  gr) ? mid : st;
      ed = (active && !gr) ? mid : ed;
    }
  }

  // ---- now consume lambda: async copy has had the whole loop to land ------
#if defined(SPU_ASYNC_LDS)
#if __has_builtin(__builtin_amdgcn_s_wait_asynccnt)
  __builtin_amdgcn_s_wait_asynccnt(0);
#else
  asm volatile("s_wait_asynccnt 0" ::: "memory");
#endif
#endif
  const float4 lam = lam_tile[tid];   // ds_load_b128, lane reads its own slot

  // ---- case A: lower >= 0 -------------------------------------------------
  const float lam_a = fmaf(width, flink(lam.x), lo);
  const float lo_a  = fmaf(2.0f * lam_a, lo, fmaf(-lam_a, lam_a, -0.5f));
  const float up_a  = up2m;

  // ---- case B: upper <= 0 -------------------------------------------------
  const float sig_u = fsig(up);
  const float lo_b  = -sig_u;
  const float lam_b = fmaf(width, flink(lam.y), lo);
  const float s_b   = fsig(lam_b);
  const float wU_b  = fmaf(s_b, s_b, -s_b);        // -s(1-s) = s^2 - s
  const float bU_b  = fmaf(-wU_b, lam_b, -s_b);
  const float up_b  = fmaf(wU_b, lo, bU_b);

  // ---- case C upper -------------------------------------------------------
  const float up_c1 = fmaxf(-sL, up2m);
  const float lam_c = fmaf(st - lo, flink(lam.w), lo);
  const float s_c   = fsig(lam_c);
  const float wU_c2 = fmaf(s_c, s_c, -s_c);
  const float bU_c2 = fmaf(-wU_c2, lam_c, -s_c);
  const float up_c2 = fmaf(wU_c2, lo, bU_c2);
  const float up_c  = use_line ? up_c1 : up_c2;

  // ---- case C lower -------------------------------------------------------
  const float l2        = flink(lam.z);
  const float neg_lower = (lo != 0.0f) ? -lo : 1.0f;
  const float wL_c1 = (1.0f - 2.0f * l2) * (sL - 0.5f)
                    * __builtin_amdgcn_rcpf(neg_lower);
  const float lo_c1 = fmaf(wL_c1, up, -0.5f);
  const float lam_l = up * fmaf(2.0f, l2, -1.0f);
  const float lo_c2 = fmaf(2.0f * lam_l, lo, fmaf(-lam_l, lam_l, -0.5f));
  const float lo_c  = (l2 < 0.5f) ? lo_c1 : lo_c2;

  // ---- select + store -----------------------------------------------------
  const float lower_out = mask_a ? lo_a : (mask_b ? lo_b : (mask_c ? lo_c : 0.0f));
  const float upper_out = mask_a ? up_a : (mask_b ? up_b : (mask_c ? up_c : 0.0f));

  if (i < n) {
    __builtin_nontemporal_store(lower_out, out + i);
    __builtin_nontemporal_store(upper_out, out + n + i);
  }
}

extern "C" void kernel_launch(void* const* d_in, const int* in_sizes, int n_in,
                              void* d_out, int out_size, void* d_ws, size_t ws_size,
                              hipStream_t stream) {
  (void)n_in; (void)out_size; (void)d_ws; (void)ws_size;
  const float*  lower  = (const float*)d_in[0];
  const float*  upper  = (const float*)d_in[1];
  const float4* lambda = (const float4*)d_in[2];
  float*        out    = (float*)d_out;
  const int n = in_sizes[0];
  const int blocks = (n + 255) / 256;
  spu_transformer_kernel<<<blocks, 256, 0, stream>>>(lower, upper, lambda, out, n);
}